// GraphSAGE_47571057770577
// MI455X (gfx1250) — compile-verified
//
#include <hip/hip_runtime.h>

// ---------------------------------------------------------------------------
// GraphSAGE (3 layers, mean aggregation) for MI455X / gfx1250, wave32.
//
//   One-time CSR build (int histogram + LDS scan + cursor scatter) turns the
//   3x edge scatter-sum (614M f32 L2 atomics) into 3x pure gather passes
//   (~0.82 GB/layer of L2-resident loads).
//
//   Per layer:  neigh[n] = (1/max(deg,1)) * sum_{e: dst=n} h[src[e]]   (gather)
//               out = relu?(h @ Wself + neigh @ Wneigh + b)            (WMMA)
// ---------------------------------------------------------------------------

typedef __attribute__((ext_vector_type(2))) float v2f;
typedef __attribute__((ext_vector_type(8))) float v8f;

#define N_NODES 50000
#define D 128
#define SCAN_B 256

// ---- degree histogram: one thread per edge (int atomics, once) ------------
__global__ void __launch_bounds__(256)
deg_hist_kernel(const int* __restrict__ dst, int* __restrict__ degi, int nE) {
  int e = blockIdx.x * blockDim.x + threadIdx.x;
  if (e < nE) atomicAdd(&degi[dst[e]], 1);
}

// ---- per-block exclusive scan (Hillis-Steele in LDS) ----------------------
__global__ void __launch_bounds__(SCAN_B)
scan_block_kernel(const int* __restrict__ in, int* __restrict__ out,
                  int* __restrict__ blockSums, int n) {
  __shared__ int tmp[SCAN_B];
  const int gid = blockIdx.x * SCAN_B + threadIdx.x;
  const int v = (gid < n) ? in[gid] : 0;
  tmp[threadIdx.x] = v;
  __syncthreads();
  for (int off = 1; off < SCAN_B; off <<= 1) {
    int t = (threadIdx.x >= off) ? tmp[threadIdx.x - off] : 0;
    __syncthreads();
    tmp[threadIdx.x] += t;
    __syncthreads();
  }
  const int incl = tmp[threadIdx.x];
  if (gid < n) out[gid] = incl - v;                    // exclusive within block
  if (threadIdx.x == SCAN_B - 1) blockSums[blockIdx.x] = incl;
}

// ---- scan of per-block totals (nb <= 256, single block, in place) ---------
__global__ void __launch_bounds__(SCAN_B)
scan_tops_kernel(int* __restrict__ sums, int nb) {
  __shared__ int tmp[SCAN_B];
  const int v = (threadIdx.x < nb) ? sums[threadIdx.x] : 0;
  tmp[threadIdx.x] = v;
  __syncthreads();
  for (int off = 1; off < SCAN_B; off <<= 1) {
    int t = (threadIdx.x >= off) ? tmp[threadIdx.x - off] : 0;
    __syncthreads();
    tmp[threadIdx.x] += t;
    __syncthreads();
  }
  if (threadIdx.x < nb) sums[threadIdx.x] = tmp[threadIdx.x] - v;  // exclusive
}

// ---- finalize: add block offsets, init scatter cursors, compute invdeg ----
__global__ void __launch_bounds__(SCAN_B)
scan_finalize_kernel(int* __restrict__ offs, const int* __restrict__ blockSums,
                     int* __restrict__ cursor, const int* __restrict__ degi,
                     float* __restrict__ invdeg, int n) {
  const int gid = blockIdx.x * SCAN_B + threadIdx.x;
  if (gid >= n) return;
  const int o = offs[gid] + blockSums[blockIdx.x];
  offs[gid] = o;
  cursor[gid] = o;
  invdeg[gid] = 1.0f / fmaxf((float)degi[gid], 1.0f);
}

// ---- CSR fill: scatter edge sources into per-dst buckets ------------------
__global__ void __launch_bounds__(256)
csr_fill_kernel(const int* __restrict__ src, const int* __restrict__ dst,
                int* __restrict__ cursor, int* __restrict__ csr, int nE) {
  int e = blockIdx.x * blockDim.x + threadIdx.x;
  if (e >= nE) return;
  const int pos = atomicAdd(&cursor[dst[e]], 1);
  csr[pos] = src[e];
}

// ---- mean aggregation: one wave32 per node, zero atomics ------------------
// Lane L owns features [4L, 4L+4). Batch 4 independent index loads per
// iteration so the index->address->feature-load latency chains overlap
// (4 independent b32 + 4 independent b128 per clause instead of a serial
// per-edge round trip). Scale by invdeg once, store the mean.
__global__ void __launch_bounds__(256)
gather_kernel(const float* __restrict__ h, const int* __restrict__ csr,
              const int* __restrict__ offs, const int* __restrict__ degi,
              const float* __restrict__ invdeg, float* __restrict__ neigh) {
  const unsigned tid  = blockIdx.x * blockDim.x + threadIdx.x;
  const unsigned node = tid >> 5;                 // one wave per node
  const unsigned lane = tid & 31;
  if (node >= N_NODES) return;
  const int start = offs[node];
  const int cnt   = degi[node];
  const float* __restrict__ hl = h + lane * 4;    // lane's feature slot
  float4 acc = {0.f, 0.f, 0.f, 0.f};

  int j = 0;
  for (; j + 4 <= cnt; j += 4) {
    const int s0 = csr[start + j + 0];            // 4 independent index loads
    const int s1 = csr[start + j + 1];
    const int s2 = csr[start + j + 2];
    const int s3 = csr[start + j + 3];
    const float4 v0 = *reinterpret_cast<const float4*>(hl + (size_t)s0 * D);
    const float4 v1 = *reinterpret_cast<const float4*>(hl + (size_t)s1 * D);
    const float4 v2 = *reinterpret_cast<const float4*>(hl + (size_t)s2 * D);
    const float4 v3 = *reinterpret_cast<const float4*>(hl + (size_t)s3 * D);
    acc.x += (v0.x + v1.x) + (v2.x + v3.x);
    acc.y += (v0.y + v1.y) + (v2.y + v3.y);
    acc.z += (v0.z + v1.z) + (v2.z + v3.z);
    acc.w += (v0.w + v1.w) + (v2.w + v3.w);
  }
  for (; j < cnt; ++j) {                          // remainder (0..3 edges)
    const int s = csr[start + j];
    const float4 v = *reinterpret_cast<const float4*>(hl + (size_t)s * D);
    acc.x += v.x; acc.y += v.y; acc.z += v.z; acc.w += v.w;
  }

  const float inv = invdeg[node];
  acc.x *= inv; acc.y *= inv; acc.z *= inv; acc.w *= inv;
  *reinterpret_cast<float4*>(neigh + (size_t)node * D + lane * 4) = acc;
}

// ---- fused dual-GEMM + bias + relu via WMMA f32 16x16x4 -------------------
// grid.x = N_NODES/16 row tiles; block = 256 threads = 8 waves, wave w owns
// column tile w. C += A_self x B_self + A_neigh x B_neigh over K=0..127/4.
// VGPR layouts per CDNA5 ISA; EXEC all-1s (grid divides exactly).
__global__ void __launch_bounds__(256)
sage_gemm_kernel(const float* __restrict__ hin, const float* __restrict__ neigh,
                 const float* __restrict__ wself, const float* __restrict__ wneigh,
                 const float* __restrict__ bias, float* __restrict__ out,
                 int do_relu) {
  const int lane    = threadIdx.x & 31;
  const int wave    = threadIdx.x >> 5;   // column tile 0..7
  const int rowTile = blockIdx.x;         // 0..3124
  const int m       = lane & 15;
  const int half    = lane >> 4;          // 0: K pair {0,1}, 1: K pair {2,3}
  const int kOff    = half * 2;

  const int row = rowTile * 16 + m;       // A row handled by this lane
  const int col = wave * 16 + m;          // B/C column handled by this lane

  const float* __restrict__ hrow = hin   + (size_t)row * D;
  const float* __restrict__ nrow = neigh + (size_t)row * D;

  v8f c = {0.f, 0.f, 0.f, 0.f, 0.f, 0.f, 0.f, 0.f};

#pragma unroll 4
  for (int k0 = 0; k0 < D; k0 += 4) {
    v2f a = *reinterpret_cast<const v2f*>(hrow + k0 + kOff);     // 8B load
    v2f b;
    b.x = wself[(size_t)(k0 + kOff + 0) * D + col];
    b.y = wself[(size_t)(k0 + kOff + 1) * D + col];
    c = __builtin_amdgcn_wmma_f32_16x16x4_f32(false, a, false, b,
                                              (short)0, c, false, false);
    v2f an = *reinterpret_cast<const v2f*>(nrow + k0 + kOff);
    v2f bn;
    bn.x = wneigh[(size_t)(k0 + kOff + 0) * D + col];
    bn.y = wneigh[(size_t)(k0 + kOff + 1) * D + col];
    c = __builtin_amdgcn_wmma_f32_16x16x4_f32(false, an, false, bn,
                                              (short)0, c, false, false);
  }

  const float bcol = bias[col];
#pragma unroll
  for (int i = 0; i < 8; ++i) {
    const int orow = rowTile * 16 + i + half * 8;  // C VGPR i -> row i / i+8
    float v = c[i] + bcol;
    if (do_relu) v = fmaxf(v, 0.f);
    out[(size_t)orow * D + col] = v;
  }
}

// ---------------------------------------------------------------------------
extern "C" void kernel_launch(void* const* d_in, const int* in_sizes, int n_in,
                              void* d_out, int out_size, void* d_ws, size_t ws_size,
                              hipStream_t stream) {
  const float* feats = (const float*)d_in[0];
  const int*   src   = (const int*)d_in[1];
  const int*   dst   = (const int*)d_in[2];
  const float* wself[3]  = {(const float*)d_in[3], (const float*)d_in[6], (const float*)d_in[9]};
  const float* wneigh[3] = {(const float*)d_in[4], (const float*)d_in[7], (const float*)d_in[10]};
  const float* bias[3]   = {(const float*)d_in[5], (const float*)d_in[8], (const float*)d_in[11]};
  const int nE = in_sizes[1];
  float* out = (float*)d_out;
  (void)ws_size; (void)n_in; (void)out_size;

  // ---- workspace carve-out (~84 MB) ----
  char* ws = (char*)d_ws;
  const size_t featBytes = (size_t)N_NODES * D * sizeof(float);
  float* h1     = (float*)ws; ws += featBytes;
  float* h2     = (float*)ws; ws += featBytes;
  float* neigh  = (float*)ws; ws += featBytes;
  int*   csr    = (int*)ws;   ws += (size_t)nE * sizeof(int);
  int*   degi   = (int*)ws;   ws += (size_t)N_NODES * sizeof(int);
  int*   offs   = (int*)ws;   ws += (size_t)N_NODES * sizeof(int);
  int*   cursor = (int*)ws;   ws += (size_t)N_NODES * sizeof(int);
  float* invdeg = (float*)ws; ws += (size_t)N_NODES * sizeof(float);
  int*   blockSums = (int*)ws; ws += 1024 * sizeof(int);

  const int nScanBlocks = (N_NODES + SCAN_B - 1) / SCAN_B;   // 196 (<256)

  // ---- CSR build (once; reused by all 3 layers) ----
  hipMemsetAsync(degi, 0, (size_t)N_NODES * sizeof(int), stream);
  deg_hist_kernel<<<(nE + 255) / 256, 256, 0, stream>>>(dst, degi, nE);
  scan_block_kernel<<<nScanBlocks, SCAN_B, 0, stream>>>(degi, offs, blockSums, N_NODES);
  scan_tops_kernel<<<1, SCAN_B, 0, stream>>>(blockSums, nScanBlocks);
  scan_finalize_kernel<<<nScanBlocks, SCAN_B, 0, stream>>>(offs, blockSums, cursor,
                                                           degi, invdeg, N_NODES);
  csr_fill_kernel<<<(nE + 255) / 256, 256, 0, stream>>>(src, dst, cursor, csr, nE);

  // ---- 3 SAGE layers ----
  const float* hin = feats;
  for (int l = 0; l < 3; ++l) {
    float* hout = (l == 0) ? h1 : (l == 1) ? h2 : out;
    gather_kernel<<<(N_NODES * 32 + 255) / 256, 256, 0, stream>>>(
        hin, csr, offs, degi, invdeg, neigh);
    sage_gemm_kernel<<<N_NODES / 16, 256, 0, stream>>>(
        hin, neigh, wself[l], wneigh[l], bias[l], hout, (l < 2) ? 1 : 0);
    hin = hout;
  }
}